// Yolo_loss_81131932221968
// MI455X (gfx1250) — compile-verified
//
#include <hip/hip_runtime.h>

// ---------------- problem constants (match reference setup_inputs) ----------
#define NBATCH 16
#define NLAB   100
#define NCLS   80
#define EPSF   1e-16f
#define K4PI2  0.40528473456935108578f   // 4 / pi^2

__constant__ float c_anch[9][2] = {
  {12.f,16.f},{19.f,36.f},{40.f,28.f},{36.f,75.f},{76.f,55.f},
  {72.f,146.f},{142.f,110.f},{192.f,243.f},{459.f,401.f}
};
__constant__ float c_stride[3] = {8.f, 16.f, 32.f};

typedef __attribute__((ext_vector_type(2))) float v2f;
typedef __attribute__((ext_vector_type(8))) float v8f;

__device__ __forceinline__ float sigmoidf_(float x) { return 1.f / (1.f + __expf(-x)); }

__device__ __forceinline__ float bcef_(float p, float t) {
  p = fminf(fmaxf(p, 1e-7f), 1.f - 1e-7f);
  return -(t * __logf(p) + (1.f - t) * __logf(1.f - p));
}

// ------------------------- sizes / workspace layout -------------------------
constexpr int kFF0 = 76 * 76, kFF1 = 38 * 38, kFF2 = 19 * 19;
constexpr int kC0 = 3 * kFF0, kC1 = 3 * kFF1, kC2 = 3 * kFF2;      // cells/image
constexpr int kMatchN = NBATCH * (kC0 + kC1 + kC2);                // 363,888
constexpr int kB0 = (kC0 + 255) / 256;                             // 68
constexpr int kB1 = (kC1 + 255) / 256;                             // 17
constexpr int kB2 = (kC2 + 255) / 256;                             // 5
constexpr int kP0 = kB0 * NBATCH, kP1 = kB1 * NBATCH, kP2 = kB2 * NBATCH;
constexpr int kNPart = kP0 + kP1 + kP2;                            // 1440
constexpr int kNPartPad = ((kNPart + 63) / 64) * 64;               // 1472 (mult of 64)

// ----------------------------- init workspace -------------------------------
// match = -1 everywhere, gates = 0, and zero the partials pad region so the
// WMMA reduction can run bounds-check-free (real slots are fully rewritten by
// the cell kernels every launch; the pad stays 0 -> deterministic).
__global__ void init_kernel(int* __restrict__ match, int nmatch,
                            int* __restrict__ gate, int ngate,
                            float* __restrict__ part_pad, int npad) {
  int i = blockIdx.x * blockDim.x + threadIdx.x;
  if (i < nmatch) match[i] = -1;
  if (i < ngate)  gate[i]  = 0;
  if (i < npad)   part_pad[i] = 0.f;
}

// ------------------- per-label targets + scatter of matches -----------------
__global__ void label_kernel(const float* __restrict__ boxes,
                             const int* __restrict__ classes,
                             float4* __restrict__ tbox,      // [3][B][M] (tx,ty,tw,th)
                             float* __restrict__ validA,     // [B][M]
                             int* __restrict__ match,        // concat of 3 scales
                             int* __restrict__ gate) {       // [3][B]
  const int b = blockIdx.x;
  const int m = threadIdx.x;
  if (m >= NLAB) return;

  const float x1 = boxes[(b * NLAB + m) * 4 + 0];
  const float y1 = boxes[(b * NLAB + m) * 4 + 1];
  const float x2 = boxes[(b * NLAB + m) * 4 + 2];
  const float y2 = boxes[(b * NLAB + m) * 4 + 3];
  const float cf = (float)classes[b * NLAB + m];
  const bool valid = (x1 + y1 + x2 + y2 + cf) > 0.f;
  validA[b * NLAB + m] = valid ? 1.f : 0.f;

  const int fs[3]   = {76, 38, 19};
  const int moff[3] = {0, NBATCH * kC0, NBATCH * (kC0 + kC1)};

  #pragma unroll
  for (int s = 0; s < 3; ++s) {
    const float inv = 1.f / c_stride[s];
    const float tx = (x1 + x2) * 0.5f * inv;
    const float ty = (y1 + y2) * 0.5f * inv;
    const float tw = (x2 - x1) * inv;
    const float th = (y2 - y1) * inv;
    tbox[(s * NBATCH + b) * NLAB + m] = make_float4(tx, ty, tw, th);

    // anchor-ratio argmax over all 9 anchors (first-max, like jnp.argmax)
    float bestr = -1e30f; int bi = 0;
    #pragma unroll
    for (int i = 0; i < 9; ++i) {
      const float agx = c_anch[i][0] * inv, agy = c_anch[i][1] * inv;
      const float inter = fminf(tw, agx) * fminf(th, agy);
      const float uni   = tw * th + agx * agy - inter + EPSF;
      const float r = inter / uni;
      if (r > bestr) { bestr = r; bi = i; }
    }
    if (valid && (bi / 3 == s)) {
      const int ai = bi % 3;
      const int ti = (int)floorf(tx), tj = (int)floorf(ty);
      const int f = fs[s];
      if (ti >= 0 && ti < f && tj >= 0 && tj < f) {
        const int cell = ai * f * f + tj * f + ti;
        atomicMax(&match[moff[s] + b * (3 * f * f) + cell], m);  // deterministic dup policy
        gate[s * NBATCH + b] = 1;   // all writers store 1 -> deterministic
      }
    }
  }
}

// ------------------------------ main per-cell loss --------------------------
template <int F, int SID>
__global__ void cell_kernel(const float* __restrict__ out,
                            const float4* __restrict__ tbox_s,   // [B][M]
                            const float* __restrict__ validA,    // [B][M]
                            const int* __restrict__ match_s,     // [B][3*F*F]
                            const int* __restrict__ gate_s,      // [B]
                            const int* __restrict__ classes,
                            float* __restrict__ partials) {
  constexpr int FF = F * F;
  constexpr int NCELL = 3 * FF;
  const int b = blockIdx.y;
  const int tid = threadIdx.x;
  const int c = blockIdx.x * 256 + tid;

  __shared__ float4 s_txyxy[NLAB];
  __shared__ float  s_area[NLAB];
  __shared__ float  s_val[NLAB];
  __shared__ float  s_red[256];

  for (int m = tid; m < NLAB; m += 256) {
    const float4 t = tbox_s[b * NLAB + m];
    s_txyxy[m] = make_float4(t.x - t.z * 0.5f, t.y - t.w * 0.5f,
                             t.x + t.z * 0.5f, t.y + t.w * 0.5f);
    s_area[m] = t.z * t.w;
    s_val[m]  = validA[b * NLAB + m];
  }
  __syncthreads();

  float loss = 0.f;
  if (c < NCELL) {
    const int a = c / FF, rem = c - a * FF;
    const int j = rem / F, i = rem - j * F;
    const float* ob = out + ((size_t)b * 255 + a * 85) * FF + rem;
    const float o0 = ob[0], o1 = ob[FF], o2 = ob[2 * FF], o3 = ob[3 * FF], o4 = ob[4 * FF];

    const float aw = c_anch[3 * SID + a][0] / c_stride[SID];
    const float ah = c_anch[3 * SID + a][1] / c_stride[SID];
    const float px = sigmoidf_(o0) + (float)i;
    const float py = sigmoidf_(o1) + (float)j;
    const float pw = __expf(o2) * aw;
    const float ph = __expf(o3) * ah;
    const float px1 = px - pw * 0.5f, py1 = py - ph * 0.5f;
    const float px2 = px + pw * 0.5f, py2 = py + ph * 0.5f;
    const float parea = pw * ph;

    // best IoU vs all valid targets (targets in LDS)
    float best = -1.f;
    #pragma unroll 4
    for (int m = 0; m < NLAB; ++m) {
      const float4 t = s_txyxy[m];
      const float w = fminf(px2, t.z) - fmaxf(px1, t.x);
      const float h = fminf(py2, t.w) - fmaxf(py1, t.y);
      const float inter = fmaxf(w, 0.f) * fmaxf(h, 0.f);
      const float iou = inter / (parea + s_area[m] - inter + EPSF);
      best = fmaxf(best, s_val[m] > 0.f ? iou : -1.f);
    }

    const int mm = match_s[b * NCELL + c];
    const float g = gate_s[b] ? 1.f : 0.f;
    const float tobj = (mm >= 0) ? 1.f : 0.f;
    const float objmask = (mm >= 0) ? 1.f : ((best < 0.5f) ? g : 0.f);
    loss += bcef_(sigmoidf_(o4), tobj) * objmask;

    if (mm >= 0) {   // class BCE + CIoU only at matched cells (mask sparsity)
      const int cls = classes[b * NLAB + mm];
      for (int k = 0; k < NCLS; ++k)
        loss += bcef_(sigmoidf_(ob[(5 + k) * FF]), (k == cls) ? 1.f : 0.f);

      const float4 g4 = tbox_s[b * NLAB + mm];
      const float gx = g4.x, gy = g4.y, gw = g4.z, gh = g4.w;
      const float ts = 2.f - gw * gh / (float)FF;
      const float gx1 = gx - gw * 0.5f, gy1 = gy - gh * 0.5f;
      const float gx2 = gx + gw * 0.5f, gy2 = gy + gh * 0.5f;
      // NOTE: replicate reference exactly (ibr uses maximum, cbr uses minimum)
      const float itlx = fmaxf(px1, gx1), itly = fmaxf(py1, gy1);
      const float ibrx = fmaxf(px2, gx2), ibry = fmaxf(py2, gy2);
      const float ctlx = fminf(px1, gx1), ctly = fminf(py1, gy1);
      const float cbrx = fminf(px2, gx2), cbry = fminf(py2, gy2);
      const float inter = (ibrx - itlx) * (ibry - itly);
      const float uni = parea + gw * gh - inter + EPSF;
      const float ious = inter / uni;
      const float cdist = (cbrx - ctlx) * (cbrx - ctlx) + (cbry - ctly) * (cbry - ctly) + EPSF;
      const float cent = (px - gx) * (px - gx) + (py - gy) * (py - gy);
      const float dv = atanf(pw / ph) - atanf(gw / gh);
      const float v = K4PI2 * dv * dv;
      const float alpha = v / (1.f - ious + v + EPSF);
      const float cious = fminf(fmaxf(ious - cent / cdist - alpha * v, -1.f), 1.f);
      loss += (1.f - cious) * ts;
    }
  }

  // deterministic block tree reduction
  s_red[tid] = loss;
  __syncthreads();
  for (int s = 128; s > 0; s >>= 1) {
    if (tid < s) s_red[tid] += s_red[tid + s];
    __syncthreads();
  }
  if (tid == 0) partials[blockIdx.y * gridDim.x + blockIdx.x] = s_red[0];
}

// ------------- final reduction: f32 WMMA (ones * data) column-sums ----------
// Partials are padded with zeros to a multiple of 64 at init time, so the loop
// has a compile-time trip count and NO bounds checks: EXEC stays all-1s and the
// body is a pure load -> v_wmma stream (WMMA requires EXEC==all ones anyway).
__global__ __launch_bounds__(32) void final_reduce(const float* __restrict__ partials,
                                                   float* __restrict__ outv) {
  const int lane = threadIdx.x;           // exactly one wave32
  v2f ones; ones.x = 1.f; ones.y = 1.f;   // A = 16x4 all-ones
  v8f acc = {};                            // C/D accumulator (f32, 16x16)
  #pragma unroll 1
  for (int base = 0; base < kNPartPad; base += 64) {
    // B = 4x16 f32: VGPR0 rows K=0,1 ; VGPR1 rows K=2,3  (64 partials/iter)
    v2f bb;
    bb.x = partials[base + lane];
    bb.y = partials[base + 32 + lane];
    acc = __builtin_amdgcn_wmma_f32_16x16x4_f32(
        /*neg_a=*/false, ones, /*neg_b=*/false, bb,
        /*c_mod=*/(short)0, acc, /*reuse_a=*/false, /*reuse_b=*/false);
  }
  // acc[0] on lane l holds column-sum s_{l&15}; fold 16 columns with shuffles
  float v = acc[0];
  v += __shfl_xor(v, 1, 16);
  v += __shfl_xor(v, 2, 16);
  v += __shfl_xor(v, 4, 16);
  v += __shfl_xor(v, 8, 16);
  if (lane == 0) outv[0] = v;
}

// --------------------------------- launcher ---------------------------------
extern "C" void kernel_launch(void* const* d_in, const int* in_sizes, int n_in,
                              void* d_out, int out_size, void* d_ws, size_t ws_size,
                              hipStream_t stream) {
  const float* out0   = (const float*)d_in[0];
  const float* out1   = (const float*)d_in[1];
  const float* out2   = (const float*)d_in[2];
  const float* boxes  = (const float*)d_in[3];
  const int*   classes= (const int*)d_in[4];
  float* outv = (float*)d_out;

  // workspace carve-up (all 16B aligned)
  char* w = (char*)d_ws;
  int* match = (int*)w;                      w += (size_t)((kMatchN * 4 + 255) / 256) * 256;
  float4* tbox = (float4*)w;                 w += (size_t)3 * NBATCH * NLAB * sizeof(float4);
  float* validA = (float*)w;                 w += (size_t)NBATCH * NLAB * sizeof(float);
  int* gate = (int*)w;                       w += 256;
  float* partials = (float*)w;

  init_kernel<<<(kMatchN + 255) / 256, 256, 0, stream>>>(
      match, kMatchN, gate, 3 * NBATCH,
      partials + kNPart, kNPartPad - kNPart);   // zero only the pad slots
  label_kernel<<<NBATCH, 128, 0, stream>>>(boxes, classes, tbox, validA, match, gate);

  cell_kernel<76, 0><<<dim3(kB0, NBATCH), 256, 0, stream>>>(
      out0, tbox + 0 * NBATCH * NLAB, validA, match, gate, classes, partials);
  cell_kernel<38, 1><<<dim3(kB1, NBATCH), 256, 0, stream>>>(
      out1, tbox + 1 * NBATCH * NLAB, validA, match + NBATCH * kC0, gate + NBATCH,
      classes, partials + kP0);
  cell_kernel<19, 2><<<dim3(kB2, NBATCH), 256, 0, stream>>>(
      out2, tbox + 2 * NBATCH * NLAB, validA, match + NBATCH * (kC0 + kC1), gate + 2 * NBATCH,
      classes, partials + kP0 + kP1);

  final_reduce<<<1, 32, 0, stream>>>(partials, outv);
}